// SAGE_56092272886197
// MI455X (gfx1250) — compile-verified
//
#include <hip/hip_runtime.h>

#define DIN 128   // D_IN == D_HID == 128 in the reference

typedef float v2f __attribute__((ext_vector_type(2)));
typedef float v8f __attribute__((ext_vector_type(8)));

// ---------------- utility kernels ----------------

__global__ void __launch_bounds__(256) zero_f32(float* __restrict__ p, long n) {
  long i = (long)blockIdx.x * blockDim.x + threadIdx.x;
  long stride = (long)gridDim.x * blockDim.x;
  for (; i < n; i += stride) p[i] = 0.0f;
}

__global__ void __launch_bounds__(256) zero_f4(float4* __restrict__ p, long n4) {
  long i = (long)blockIdx.x * blockDim.x + threadIdx.x;
  long stride = (long)gridDim.x * blockDim.x;
  float4 z; z.x = z.y = z.z = z.w = 0.0f;
  for (; i < n4; i += stride) p[i] = z;
}

__global__ void __launch_bounds__(256) degree_k(const int* __restrict__ dst,
                                                float* __restrict__ deg, int E) {
  int i = blockIdx.x * blockDim.x + threadIdx.x;
  if (i < E) atomicAdd(&deg[dst[i]], 1.0f);
}

__global__ void __launch_bounds__(256) invdeg_k(float* __restrict__ deg, int N) {
  int i = blockIdx.x * blockDim.x + threadIdx.x;
  if (i < N) deg[i] = 1.0f / fmaxf(deg[i], 1.0f);
}

// ---------------- edge scatter: one wave32 per edge ----------------
// lane l moves feature channels [4l, 4l+4): b128 load + 4 f32 atomic adds.
// Feature table + accumulator are L2-resident (51.2 MB each vs 192 MB L2).

__global__ void __launch_bounds__(256) scatter_k(const float* __restrict__ h,
                                                 const int* __restrict__ src,
                                                 const int* __restrict__ dst,
                                                 float* __restrict__ agg, int E) {
  int gw   = (blockIdx.x * blockDim.x + threadIdx.x) >> 5;  // global wave id
  int lane = threadIdx.x & 31;
  int nw   = (gridDim.x * blockDim.x) >> 5;
  for (int e = gw; e < E; e += nw) {
    int s = src[e];                      // wave-uniform -> scalarized
    int d = dst[e];
    const float4 v = *(const float4*)(h + (size_t)s * DIN + lane * 4);
    float* p = agg + (size_t)d * DIN + lane * 4;
    atomicAdd(p + 0, v.x);
    atomicAdd(p + 1, v.y);
    atomicAdd(p + 2, v.z);
    atomicAdd(p + 3, v.w);
  }
}

// ---------------- fused SAGE layer GEMM ----------------
// out[m,:] = act( (agg[m,:]*invdeg[m]) @ Wl + h[m,:] @ Wr + bias )
//
// block = 256 threads (8 waves), M-tile = 32 rows. Wave w owns output
// columns [16w, 16w+16) and computes TWO stacked 16x16 tiles (rows m0..m0+15
// and m0+16..m0+31) that share each B fragment -> weight loads reused 2x.
// All weight loads use a hoisted per-lane base pointer + compile-time
// immediate offsets (k*DOUT*4 <= 63488 B fits the 24-bit IOFFSET).
//
// V_WMMA_F32_16X16X4_F32 fragment layout:
//  A 16x4: lanes 0-15 -> M=lane, VGPR0/1 = K=0/1; lanes 16-31 -> K=2/3.
//  B 4x16: mirrored (row striped across lanes per VGPR).
//  C/D 16x16: VGPR r, lanes 0-15 -> (M=r, N=lane); lanes 16-31 -> (M=r+8).

template <int DOUT, bool RELU>
__global__ void __launch_bounds__(256) sage_gemm_k(
    const float* __restrict__ agg, const float* __restrict__ h,
    const float* __restrict__ invdeg,
    const float* __restrict__ Wl, const float* __restrict__ Wr,
    const float* __restrict__ bias, float* __restrict__ out, int N)
{
  // +4 float pad -> row stride 132: wave32 64-bit LDS reads hit all 64 banks once
  __shared__ float sA[32][DIN + 4];
  __shared__ float sH[32][DIN + 4];

  const int m0 = blockIdx.x * 32;

  // cooperative stage: 32x128 of (agg * 1/deg) and 32x128 of h
  for (int idx = threadIdx.x; idx < 32 * (DIN / 4); idx += 256) {
    int row = idx >> 5;              // DIN/4 == 32 float4 per row
    int c4  = (idx & 31) * 4;
    int g   = m0 + row;
    float4 va; va.x = va.y = va.z = va.w = 0.0f;
    float4 vh = va;
    if (g < N) {
      va = *(const float4*)(agg + (size_t)g * DIN + c4);
      vh = *(const float4*)(h   + (size_t)g * DIN + c4);
      float s = invdeg[g];
      va.x *= s; va.y *= s; va.z *= s; va.w *= s;
    }
    *(float4*)&sA[row][c4] = va;
    *(float4*)&sH[row][c4] = vh;
  }
  __syncthreads();

  const int wave = threadIdx.x >> 5;
  const int lane = threadIdx.x & 31;
  const int half = lane >> 4;
  const int lrow = lane & 15;
  const int n0   = wave * 16;
  if (n0 >= DOUT) return;            // wave-uniform; EXEC all-ones for WMMA path

  // hoisted per-lane bases: loop offsets become compile-time immediates
  const float* wl  = Wl + (size_t)(2 * half) * DOUT + n0 + lrow;
  const float* wr  = Wr + (size_t)(2 * half) * DOUT + n0 + lrow;
  const float* pa0 = &sA[lrow][2 * half];
  const float* pa1 = &sA[16 + lrow][2 * half];
  const float* ph0 = &sH[lrow][2 * half];
  const float* ph1 = &sH[16 + lrow][2 * half];

  v8f acc0, acc1;
  const float bv = bias[n0 + lrow];  // bias depends on column only
#pragma unroll
  for (int r = 0; r < 8; ++r) { acc0[r] = bv; acc1[r] = bv; }

#pragma unroll
  for (int k = 0; k < DIN; k += 4) {
    v2f bl;                          // B frag of Wl, shared by both M tiles
    bl.x = wl[k * DOUT];
    bl.y = wl[k * DOUT + DOUT];
    v2f a0 = *(const v2f*)(pa0 + k);
    v2f a1 = *(const v2f*)(pa1 + k);
    acc0 = __builtin_amdgcn_wmma_f32_16x16x4_f32(false, a0, false, bl,
                                                 (short)0, acc0, false, false);
    acc1 = __builtin_amdgcn_wmma_f32_16x16x4_f32(false, a1, false, bl,
                                                 (short)0, acc1, false, false);

    v2f br;                          // B frag of Wr, shared by both M tiles
    br.x = wr[k * DOUT];
    br.y = wr[k * DOUT + DOUT];
    v2f h0 = *(const v2f*)(ph0 + k);
    v2f h1 = *(const v2f*)(ph1 + k);
    acc0 = __builtin_amdgcn_wmma_f32_16x16x4_f32(false, h0, false, br,
                                                 (short)0, acc0, false, false);
    acc1 = __builtin_amdgcn_wmma_f32_16x16x4_f32(false, h1, false, br,
                                                 (short)0, acc1, false, false);
  }

#pragma unroll
  for (int r = 0; r < 8; ++r) {
    int row0 = m0 + r + 8 * half;
    int row1 = row0 + 16;
    float v0 = acc0[r];
    float v1 = acc1[r];
    if (RELU) { v0 = fmaxf(v0, 0.0f); v1 = fmaxf(v1, 0.0f); }
    if (row0 < N) out[(size_t)row0 * DOUT + n0 + lrow] = v0;
    if (row1 < N) out[(size_t)row1 * DOUT + n0 + lrow] = v1;
  }
}

// ---------------- host glue ----------------

extern "C" void kernel_launch(void* const* d_in, const int* in_sizes, int n_in,
                              void* d_out, int out_size, void* d_ws, size_t ws_size,
                              hipStream_t stream) {
  const float* x   = (const float*)d_in[0];
  const int*  edge = (const int*)d_in[1];
  const float* Wl0 = (const float*)d_in[2];
  const float* Wr0 = (const float*)d_in[3];
  const float* b0  = (const float*)d_in[4];
  const float* Wl1 = (const float*)d_in[5];
  const float* Wr1 = (const float*)d_in[6];
  const float* b1  = (const float*)d_in[7];
  const float* Wl2 = (const float*)d_in[8];
  const float* Wr2 = (const float*)d_in[9];
  const float* b2  = (const float*)d_in[10];
  float* out = (float*)d_out;

  const int N = in_sizes[0] / DIN;
  const int E = in_sizes[1] / 2;
  const int* src = edge;
  const int* dst = edge + E;

  // workspace: deg | agg | h   (~103 MB)
  char* ws = (char*)d_ws;
  size_t off = 0;
  float* deg  = (float*)(ws + off); off += ((size_t)N * 4 + 255) & ~(size_t)255;
  float* agg  = (float*)(ws + off); off += ((size_t)N * DIN * 4 + 255) & ~(size_t)255;
  float* hbuf = (float*)(ws + off);
  (void)ws_size; (void)n_in; (void)out_size;

  const int blk = 256;
  const long aggn4 = (long)N * DIN / 4;
  const int zgrid = (int)((aggn4 + blk - 1) / blk);
  const int sgrid = (E + 7) / 8;          // 8 waves/block, 1 edge/wave
  const int ggrid = (N + 31) / 32;        // 32 rows per block

  // degrees (once; inverted in place by invdeg_k; 1/deg scaling folded into GEMM staging)
  zero_f32<<<(N + blk - 1) / blk, blk, 0, stream>>>(deg, N);
  degree_k<<<(E + blk - 1) / blk, blk, 0, stream>>>(dst, deg, E);
  invdeg_k<<<(N + blk - 1) / blk, blk, 0, stream>>>(deg, N);

  // layer 0: x -> h (ReLU)
  zero_f4<<<zgrid, blk, 0, stream>>>((float4*)agg, aggn4);
  scatter_k<<<sgrid, blk, 0, stream>>>(x, src, dst, agg, E);
  sage_gemm_k<DIN, true><<<ggrid, blk, 0, stream>>>(agg, x, deg, Wl0, Wr0, b0, hbuf, N);

  // layer 1: h -> h (ReLU; in-place safe: each block reads exactly the rows it writes)
  zero_f4<<<zgrid, blk, 0, stream>>>((float4*)agg, aggn4);
  scatter_k<<<sgrid, blk, 0, stream>>>(hbuf, src, dst, agg, E);
  sage_gemm_k<DIN, true><<<ggrid, blk, 0, stream>>>(agg, hbuf, deg, Wl1, Wr1, b1, hbuf, N);

  // layer 2: h -> out (no activation, DOUT=64 -> waves 4..7 idle)
  zero_f4<<<zgrid, blk, 0, stream>>>((float4*)agg, aggn4);
  scatter_k<<<sgrid, blk, 0, stream>>>(hbuf, src, dst, agg, E);
  sage_gemm_k<64, false><<<ggrid, blk, 0, stream>>>(agg, hbuf, deg, Wl2, Wr2, b2, out, N);
}